// IPSLoss_24163486007858
// MI455X (gfx1250) — compile-verified
//
#include <hip/hip_runtime.h>

typedef __attribute__((ext_vector_type(2))) float v2f;
typedef __attribute__((ext_vector_type(8))) float v8f;

// Cross-lane wave32 reduction using V_WMMA_F32_16X16X4_F32.
// A-matrix layout (16x4 f32): lanes 0-15 VGPR0 = A[m][0], lanes 16-31 VGPR0 = A[m][2];
// VGPR1 = K=1 / K=3 (set to 0). With B = ones(4x16):
//   D[m][n] = acc[m] + acc[m+16]  (replicated over n)
// C/D layout: lane n in 0-15 VGPR v = D[v][n]; lane n in 16-31 VGPR v = D[v+8][n-16].
// So sum of lane0's 8 D regs covers acc[0..7]+acc[16..23], lane16 covers acc[8..15]+acc[24..31].
// readlane(s,0) + readlane(s,16) == full 32-lane sum.
// NOTE: EXEC must be all 1s at the call site (callers reconverge before calling).
__device__ __forceinline__ float wave_reduce_wmma(float acc) {
    v2f a; a.x = acc;  a.y = 0.0f;
    v2f b; b.x = 1.0f; b.y = 1.0f;
    v8f c = {};
    v8f d = __builtin_amdgcn_wmma_f32_16x16x4_f32(
        /*neg_a=*/false, a, /*neg_b=*/false, b,
        /*c_mod=*/(short)0, c, /*reuse_a=*/false, /*reuse_b=*/false);
    float s = d[0] + d[1] + d[2] + d[3] + d[4] + d[5] + d[6] + d[7];
    float s0  = __int_as_float(__builtin_amdgcn_readlane(__float_as_int(s), 0));
    float s16 = __int_as_float(__builtin_amdgcn_readlane(__float_as_int(s), 16));
    return s0 + s16; // uniform across the wave
}

#define NLABELS 5

// Pass 1: streamed, vectorized, memory-bound main loop.
// Each block writes exactly one deterministic partial sum to d_ws[blockIdx.x].
__global__ void ips_partial_kernel(const float4* __restrict__ out4,
                                   const int4*  __restrict__ lab4,
                                   const int4*  __restrict__ it4,
                                   const float* __restrict__ ip,
                                   float* __restrict__ partial,
                                   int nvec4, long long n_total) {
    const int gid    = blockIdx.x * blockDim.x + threadIdx.x;
    const int stride = gridDim.x * blockDim.x;

    float acc = 0.0f;
    for (int i = gid; i < nvec4; i += stride) {
        float4 o = out4[i];          // global_load_b128
        int4   l = lab4[i];          // global_load_b128
        int4   t = it4[i];           // global_load_b128
        // L2/WGP$-resident gathers into the 2MB propensity table
        float w0 = ip[t.x * NLABELS + l.x];
        float w1 = ip[t.y * NLABELS + l.y];
        float w2 = ip[t.z * NLABELS + l.z];
        float w3 = ip[t.w * NLABELS + l.w];
        float d0 = o.x - (float)l.x;
        float d1 = o.y - (float)l.y;
        float d2 = o.z - (float)l.z;
        float d3 = o.w - (float)l.w;
        acc = fmaf(d0 * d0, w0, acc);
        acc = fmaf(d1 * d1, w1, acc);
        acc = fmaf(d2 * d2, w2, acc);
        acc = fmaf(d3 * d3, w3, acc);
    }

    // Scalar tail (empty when N % 4 == 0, e.g. N = 2^24). Reconverges before WMMA.
    if (gid == 0) {
        const float* o = (const float*)out4;
        const int*   l = (const int*)lab4;
        const int*   t = (const int*)it4;
        for (long long i = (long long)nvec4 * 4; i < n_total; ++i) {
            float w = ip[t[i] * NLABELS + l[i]];
            float d = o[i] - (float)l[i];
            acc = fmaf(d * d, w, acc);
        }
    }

    // Intra-wave reduction via WMMA (EXEC all-ones here).
    float wt = wave_reduce_wmma(acc);

    __shared__ float wsum[32];
    const int lane = threadIdx.x & 31;
    const int wid  = threadIdx.x >> 5;
    if (lane == 0) wsum[wid] = wt;
    __syncthreads();
    if (threadIdx.x == 0) {
        const int nw = blockDim.x >> 5;
        float bsum = 0.0f;
        for (int w = 0; w < nw; ++w) bsum += wsum[w]; // fixed order: deterministic
        partial[blockIdx.x] = bsum;
    }
}

// Pass 2: one block folds the per-block partials in a fixed order.
__global__ void ips_final_kernel(const float* __restrict__ partial, int nparts,
                                 float* __restrict__ out) {
    float acc = 0.0f;
    for (int i = threadIdx.x; i < nparts; i += blockDim.x) acc += partial[i];
    // Reconverged here; EXEC all-ones for WMMA.
    float wt = wave_reduce_wmma(acc);

    __shared__ float wsum[32];
    const int lane = threadIdx.x & 31;
    const int wid  = threadIdx.x >> 5;
    if (lane == 0) wsum[wid] = wt;
    __syncthreads();
    if (threadIdx.x == 0) {
        const int nw = blockDim.x >> 5;
        float total = 0.0f;
        for (int w = 0; w < nw; ++w) total += wsum[w];
        out[0] = total;
    }
}

extern "C" void kernel_launch(void* const* d_in, const int* in_sizes, int n_in,
                              void* d_out, int out_size, void* d_ws, size_t ws_size,
                              hipStream_t stream) {
    // setup_inputs order: output (f32[N]), label (i32[N]), inverse_propensity
    // (f32[100000*5]), item (i32[N]).
    const float* output = (const float*)d_in[0];
    const int*   label  = (const int*)d_in[1];
    const float* ip     = (const float*)d_in[2];
    const int*   item   = (const int*)d_in[3];

    const long long n = (long long)in_sizes[0];
    const int nvec4 = (int)(n / 4);

    const int BLOCK = 256; // 8 wave32s per block
    int nblocks = 4096;    // 2^24 / (256*16) -> 4 vec4 iters/thread
    // Clamp to available workspace (one float partial per block).
    if ((size_t)nblocks * sizeof(float) > ws_size) {
        nblocks = (int)(ws_size / sizeof(float));
        if (nblocks < 1) nblocks = 1;
    }
    float* partial = (float*)d_ws;

    ips_partial_kernel<<<nblocks, BLOCK, 0, stream>>>(
        (const float4*)output, (const int4*)label, (const int4*)item, ip,
        partial, nvec4, n);
    ips_final_kernel<<<1, BLOCK, 0, stream>>>(partial, nblocks, (float*)d_out);
}